// DigitCaps_73744588472848
// MI455X (gfx1250) — compile-verified
//
#include <hip/hip_runtime.h>

// DigitCaps dynamic routing for MI455X (gfx1250, wave32).
// Kernel 1: u_hat = einsum('bid,iokd->biok') via V_WMMA_F32_16X16X4_F32 (fp32 in,
//           fp32 accumulate, bf16 store -> u_hat is 94 MB and L2-resident).
// Kernel 2: all 3 routing iterations fused, one workgroup per batch element;
//           b_logits (46 KB) lives entirely in LDS; one u_hat sweep per iteration.

typedef __attribute__((ext_vector_type(2))) float v2f;
typedef __attribute__((ext_vector_type(8))) float v8f;

#define B_   256
#define NI   1152
#define NO   10
#define KD   16
#define DI   8
#define NOK  160   // NO*KD

__device__ __forceinline__ unsigned short f32_to_bf16(float f) {
    union { float f; unsigned u; } c; c.f = f;
    unsigned r = c.u + 0x7FFFu + ((c.u >> 16) & 1u);   // round-to-nearest-even
    return (unsigned short)(r >> 16);
}
__device__ __forceinline__ float bf16_to_f32(unsigned short h) {
    union { unsigned u; float f; } c; c.u = ((unsigned)h) << 16;
    return c.f;
}

// ---------------------------------------------------------------------------
// Kernel 1: u_hat via fp32 WMMA.
// grid = (NI, B/16); block = 320 (10 waves; wave w handles ok-tile w).
// Per wave: C[16b x 16ok] = A[16b x 8d] * B[8d x 16ok] as two 16x16x4 WMMAs.
// A 16x4 layout: lanes 0-15 hold (M=r, K=0|1) in v0|v1; lanes 16-31 (M=r, K=2|3).
// B  4x16 layout: lanes 0-15 hold (K=0|1, N=r);         lanes 16-31 (K=2|3, N=r).
// ---------------------------------------------------------------------------
__global__ __launch_bounds__(320) void uhat_wmma_kernel(
    const float* __restrict__ x,          // [B, NI, DI]
    const float* __restrict__ W,          // [NI, NO, KD, DI] == [NI, NOK, DI]
    unsigned short* __restrict__ uh)      // [B, NI, NOK] bf16
{
    __shared__ alignas(16) unsigned short tile[16 * NOK]; // 5 KB staging
    const int i    = blockIdx.x;          // input capsule
    const int bt   = blockIdx.y << 4;     // batch tile base
    const int tid  = threadIdx.x;
    const int wave = tid >> 5;            // 0..9 -> ok tile
    const int lane = tid & 31;
    const int r    = lane & 15;
    const int hi   = lane >> 4;           // 0/1 -> K offset 0/2
    const int okb  = wave << 4;

    const float* xa = x + ((size_t)(bt + r) * NI + i) * DI + hi * 2;
    const float* wb = W + ((size_t)i * NOK + okb + r) * DI + hi * 2;
    v2f a0 = *(const v2f*)(xa);           // K = hi*2 + {0,1}
    v2f a1 = *(const v2f*)(xa + 4);       // K = 4 + hi*2 + {0,1}
    v2f b0 = *(const v2f*)(wb);
    v2f b1 = *(const v2f*)(wb + 4);

    v8f c = {};
    c = __builtin_amdgcn_wmma_f32_16x16x4_f32(false, a0, false, b0,
                                              (short)0, c, false, false);
    c = __builtin_amdgcn_wmma_f32_16x16x4_f32(false, a1, false, b1,
                                              (short)0, c, false, false);

    // C/D layout: VGPR j, lanes 0-15 -> (M=j, N=r); lanes 16-31 -> (M=j+8, N=r)
    #pragma unroll
    for (int j = 0; j < 8; ++j) {
        int m = j + (hi << 3);
        tile[m * NOK + okb + r] = f32_to_bf16(c[j]);
    }
    __syncthreads();

    // Coalesced store: 16 rows x 160 bf16 = 320 x 16B chunks, one per thread.
    const int row = tid / 20;
    const int ch  = tid % 20;
    const uint4* src = (const uint4*)(tile + row * NOK) + ch;
    uint4* dst = (uint4*)(uh + ((size_t)(bt + row) * NI + i) * NOK) + ch;
    *dst = *src;
}

// ---------------------------------------------------------------------------
// Kernel 2: fused 3-iteration dynamic routing, one workgroup per batch b.
// 256 threads = 32 groups of 8 lanes; group G owns capsule i = t*32+G;
// lane l8 in group owns the k-pair (2*l8, 2*l8+1) -> one bf16x2 dword load.
// Per iteration, ONE sweep over u_hat[b] computes agreement (vs previous v),
// the logit update + softmax, and the c-weighted s accumulation.
// ---------------------------------------------------------------------------
__global__ __launch_bounds__(256) void routing_kernel(
    const unsigned short* __restrict__ uh,   // [B, NI, NOK] bf16
    const float* __restrict__ bias,          // [NO, KD]
    float* __restrict__ vout)                // [B, NOK]
{
    __shared__ float blog[NI * NO];     // 46080 B  routing logits (whole state)
    __shared__ float spart[32 * NOK];   // 20480 B  per-group s partials
    __shared__ float sbuf[NOK];
    __shared__ float vbuf[NOK];

    const int b    = blockIdx.x;
    const int tid  = threadIdx.x;
    const int wave = tid >> 5;
    const int lane = tid & 31;
    const int G    = (wave << 2) | (lane >> 3);  // 0..31 capsule group
    const int l8   = lane & 7;                   // k-pair index

    for (int idx = tid; idx < NI * NO; idx += 256) blog[idx] = 0.0f;
    __syncthreads();

    float vreg[NO][2];
    #pragma unroll
    for (int o = 0; o < NO; ++o) { vreg[o][0] = 0.f; vreg[o][1] = 0.f; }

    for (int iter = 0; iter < 3; ++iter) {
        float sl[NO][2];
        #pragma unroll
        for (int o = 0; o < NO; ++o) { sl[o][0] = 0.f; sl[o][1] = 0.f; }

        for (int t = 0; t < NI / 32; ++t) {
            const int i = t * 32 + G;
            const unsigned* up =
                (const unsigned*)(uh + ((size_t)b * NI + i) * NOK);
            float u[NO][2];
            #pragma unroll
            for (int o = 0; o < NO; ++o) {
                unsigned pk = up[o * 8 + l8];
                u[o][0] = bf16_to_f32((unsigned short)(pk & 0xffffu));
                u[o][1] = bf16_to_f32((unsigned short)(pk >> 16));
            }
            float cc[NO];
            if (iter == 0) {
                #pragma unroll
                for (int o = 0; o < NO; ++o) cc[o] = 0.1f;   // softmax(0)
            } else {
                float bl[NO];
                #pragma unroll
                for (int o = 0; o < NO; ++o) {
                    float ag = u[o][0] * vreg[o][0] + u[o][1] * vreg[o][1];
                    ag += __shfl_xor(ag, 1, 8);              // reduce over 16 k
                    ag += __shfl_xor(ag, 2, 8);
                    ag += __shfl_xor(ag, 4, 8);
                    bl[o] = blog[i * NO + o] + ag;
                }
                if (l8 == 0) {                               // persist logits
                    #pragma unroll
                    for (int o = 0; o < NO; ++o) blog[i * NO + o] = bl[o];
                }
                float mx = bl[0];
                #pragma unroll
                for (int o = 1; o < NO; ++o) mx = fmaxf(mx, bl[o]);
                float sum = 0.f;
                #pragma unroll
                for (int o = 0; o < NO; ++o) { cc[o] = __expf(bl[o] - mx); sum += cc[o]; }
                float inv = 1.0f / sum;
                #pragma unroll
                for (int o = 0; o < NO; ++o) cc[o] *= inv;
            }
            #pragma unroll
            for (int o = 0; o < NO; ++o) {
                sl[o][0] += cc[o] * u[o][0];
                sl[o][1] += cc[o] * u[o][1];
            }
        }

        __syncthreads();
        #pragma unroll
        for (int o = 0; o < NO; ++o) {
            spart[G * NOK + o * KD + 2 * l8]     = sl[o][0];
            spart[G * NOK + o * KD + 2 * l8 + 1] = sl[o][1];
        }
        __syncthreads();
        if (tid < NOK) {                       // s = sum over groups + bias
            float s = bias[tid];
            #pragma unroll
            for (int g = 0; g < 32; ++g) s += spart[g * NOK + tid];
            sbuf[tid] = s;
        }
        __syncthreads();
        if (tid < NOK) {                       // squash
            const int o = tid >> 4;
            float sq = 0.f;
            #pragma unroll
            for (int k = 0; k < KD; ++k) { float d = sbuf[o * KD + k]; sq += d * d; }
            float scale = sq / (1.0f + sq) * __frsqrt_rn(sq + 1e-9f);
            vbuf[tid] = scale * sbuf[tid];
        }
        __syncthreads();
        #pragma unroll
        for (int o = 0; o < NO; ++o) {         // broadcast v for next sweep
            vreg[o][0] = vbuf[o * KD + 2 * l8];
            vreg[o][1] = vbuf[o * KD + 2 * l8 + 1];
        }
        __syncthreads();
    }

    if (tid < NOK) vout[(size_t)b * NOK + tid] = vbuf[tid];
}

// ---------------------------------------------------------------------------
extern "C" void kernel_launch(void* const* d_in, const int* in_sizes, int n_in,
                              void* d_out, int out_size, void* d_ws, size_t ws_size,
                              hipStream_t stream) {
    const float* x    = (const float*)d_in[0];   // [256, 1152, 8]
    const float* W    = (const float*)d_in[1];   // [1152, 10, 16, 8]
    const float* bias = (const float*)d_in[2];   // [10, 16]
    float* out = (float*)d_out;                  // [256, 10, 16]

    // Workspace: bf16 u_hat [B, NI, NOK] = 94,371,840 bytes.
    unsigned short* uh = (unsigned short*)d_ws;

    dim3 g1(NI, B_ / 16);
    uhat_wmma_kernel<<<g1, 320, 0, stream>>>(x, W, uh);
    routing_kernel<<<B_, 256, 0, stream>>>(uh, bias, out);
}